// FluidFlow_33517924778017
// MI455X (gfx1250) — compile-verified
//
#include <hip/hip_runtime.h>
#include <cmath>

// FluidFlow: 20 fused pointwise steps of g -> g + (2/(W(z)+1)-1)*dt, W via Halley.
// Compute-bound on the TRANS pipe (~22.6G transcendental ops vs 302 MB of HBM
// traffic), so: full temporal fusion in registers, hardware v_exp/v_log/v_sqrt/
// v_rcp, one-divide Halley, 4 independent chains per thread for TRANS ILP.

#define C1_F 0.5f
#define DT_F 0.01f
#define HALLEY_ITERS 12

// Native clang vector type: required by __builtin_nontemporal_* and lowers to
// 128-bit global_load_b128 / global_store_b128.
typedef float v4f __attribute__((ext_vector_type(4)));

__device__ __forceinline__ float fast_exp(float x) {
  // exp(x) = exp2(x * log2(e)) -> v_exp_f32
  return __builtin_amdgcn_exp2f(x * 1.44269504088896340736f);
}
__device__ __forceinline__ float fast_log(float x) {
  // log(x) = log2(x) * ln(2) -> v_log_f32
  return __builtin_amdgcn_logf(x) * 0.693147180559945309417f;
}

__device__ __forceinline__ float step_elem(float g) {
  // xi = sqrt(2*(log(g+1) - c1 + 1))
  float t  = fast_log(g + 1.0f);
  float xi = __builtin_amdgcn_sqrtf(2.0f * (t + (1.0f - C1_F)));
  // z = exp(xi^2/2 + c1 - 1)
  float z  = fast_exp(fmaf(xi * xi, 0.5f, C1_F - 1.0f));
  // Lambert W, w0 = log1p(z)
  float w = fast_log(z + 1.0f);
#pragma unroll
  for (int i = 0; i < HALLEY_ITERS; ++i) {
    float ew  = fast_exp(w);                      // v_exp_f32
    float f   = fmaf(w, ew, -z);                  // w*e^w - z
    float a   = w + 1.0f;
    float b   = a + a;                            // 2w + 2
    // w -= f / (ew*(w+1) - (w+2)*f/(2w+2))
    //    = w - f*(2w+2) / (ew*(w+1)*(2w+2) - (w+2)*f)   [single rcp]
    float den = fmaf(ew * a, b, -((w + 2.0f) * f));
    w = fmaf(-(f * b), __builtin_amdgcn_rcpf(den), w);
  }
  float upd = fmaf(2.0f, __builtin_amdgcn_rcpf(w + 1.0f), -1.0f);
  return fmaf(upd, DT_F, g);
}

__global__ __launch_bounds__(256) void FluidFlow_33517924778017_kernel(
    const float* __restrict__ gin, float* __restrict__ gout,
    const int* __restrict__ steps_ptr, int N) {
  // One v4f (4 columns of one row) per thread; N % 4 == 0 (N = 6144).
  const int col = ((int)blockIdx.x * (int)blockDim.x + (int)threadIdx.x) * 4;
  if (col >= N) return;
  const int row = (int)blockIdx.y;
  const long long base = (long long)row * (long long)N + (long long)col;

  __builtin_prefetch(gin + base, 0, 3);           // global_prefetch_b8

  const int steps = steps_ptr[0];                 // uniform scalar load

  const v4f v = *(const v4f*)(gin + base);        // global_load_b128
  float a0 = v.x, a1 = v.y, a2 = v.z, a3 = v.w;

  for (int s = 0; s < steps; ++s) {
    // 4 independent Halley chains -> ILP across the quarter-rate TRANS pipe
    a0 = step_elem(a0);
    a1 = step_elem(a1);
    a2 = step_elem(a2);
    a3 = step_elem(a3);
  }

  // Boundary mask collapses to: boundary cells are exactly 0 in the output.
  if ((row == 0) | (row == N - 1)) { a0 = 0.0f; a1 = 0.0f; a2 = 0.0f; a3 = 0.0f; }
  if (col == 0)     a0 = 0.0f;
  if (col + 4 == N) a3 = 0.0f;

  v4f o; o.x = a0; o.y = a1; o.z = a2; o.w = a3;
  __builtin_nontemporal_store(o, (v4f*)(gout + base));  // NT global_store_b128
}

extern "C" void kernel_launch(void* const* d_in, const int* in_sizes, int n_in,
                              void* d_out, int out_size, void* d_ws, size_t ws_size,
                              hipStream_t stream) {
  const float* gin       = (const float*)d_in[0];
  const int*   steps_ptr = (const int*)d_in[1];
  float*       gout      = (float*)d_out;

  const long long total = (long long)in_sizes[0];
  const int N = (int)llround(sqrt((double)total));   // 6144

  const int block = 256;
  const int colsPerBlock = block * 4;
  dim3 grid((unsigned)((N + colsPerBlock - 1) / colsPerBlock), (unsigned)N);
  FluidFlow_33517924778017_kernel<<<grid, dim3(block), 0, stream>>>(gin, gout, steps_ptr, N);
}